// SelfAttention_8959301779621
// MI455X (gfx1250) — compile-verified
//
#include <hip/hip_runtime.h>

// ---------------------------------------------------------------------------
// SelfAttention (degenerate einsum): out[n,q,j] = (sum_v vsum[n,v]*Wo[j,v]) + bo[j]
//   vsum[n,v] = values[n,v,:] . colsum(Wv) + sum(bv)
//   (softmax row-sum == 1 -> q/k projections and energy GEMM contribute x1.0)
// Bandwidth-bound: ~168 MB traffic -> ~7 us @ 23.3 TB/s.
// Surviving GEMM (vsum @ Wo^T) runs on V_WMMA_F32_16X16X4_F32 with B staged
// through LDS via global_load_async_to_lds_b128 (double-buffered, ASYNCcnt).
// ---------------------------------------------------------------------------

#define EMB 2048
#define NBATCH 4
#define LSEQ 2048
#define KT 64            // K-tile staged in LDS per wave
#define KTP (KT + 4)     // padded row stride: 272B keeps 16B align, banks 4*n
#define WPB 4            // waves per block in the GEMM kernel

#if defined(__gfx1250__) && __has_builtin(__builtin_amdgcn_wmma_f32_16x16x4_f32)
  #define DEVICE_FAST 1
#else
  #define DEVICE_FAST 0
#endif

typedef __attribute__((ext_vector_type(2))) float v2f;
typedef __attribute__((ext_vector_type(4))) float v4f;
typedef __attribute__((ext_vector_type(8))) float v8f;

// K0: zero rows 4..15 of the padded A matrix (so the WMMA loop needs no
// predication and EXEC stays uniform through the hot loop).
__global__ void pad_zero_kernel(float* __restrict__ vsump) {
  const int i = blockIdx.x * blockDim.x + threadIdx.x;  // 0 .. 12*EMB-1
  vsump[(size_t)NBATCH * EMB + i] = 0.f;
}

// K1: wcol[c] = sum_r Wv[r,c];  bvsum = sum(bv)  (LDS tree reduction)
__global__ void colsum_bv_kernel(const float* __restrict__ Wv,
                                 const float* __restrict__ bv,
                                 float* __restrict__ wcol,
                                 float* __restrict__ bvsum) {
  const int c = blockIdx.x * blockDim.x + threadIdx.x;   // 0..EMB-1
  float s = 0.f;
  for (int r = 0; r < EMB; ++r) s += Wv[(size_t)r * EMB + c];  // coalesced in c
  wcol[c] = s;

  __shared__ float red[256];
  if (blockIdx.x == 0) {
    float t = 0.f;
    for (int r = threadIdx.x; r < EMB; r += 256) t += bv[r];
    red[threadIdx.x] = t;
    __syncthreads();
    for (int off = 128; off > 0; off >>= 1) {
      if ((int)threadIdx.x < off) red[threadIdx.x] += red[threadIdx.x + off];
      __syncthreads();
    }
    if (threadIdx.x == 0) bvsum[0] = red[0];
  }
}

// K2: vsump[row] = values[row,:] . wcol + bvsum  (one wave32 per row, b128 loads)
__global__ void vsum_kernel(const float* __restrict__ values,
                            const float* __restrict__ wcol,
                            const float* __restrict__ bvsum,
                            float* __restrict__ vsump) {
  const int row  = blockIdx.x * (blockDim.x >> 5) + (threadIdx.x >> 5);
  const int lane = threadIdx.x & 31;
  const v4f* __restrict__ vp = (const v4f*)(values + (size_t)row * EMB);
  const v4f* __restrict__ wp = (const v4f*)wcol;
  float s = 0.f;
  for (int c = lane; c < EMB / 4; c += 32) {
    v4f a = vp[c], w = wp[c];
    s += a.x * w.x + a.y * w.y + a.z * w.z + a.w * w.w;
  }
  for (int off = 16; off > 0; off >>= 1)   // wave32 butterfly reduce
    s += __shfl_xor(s, off, 32);
  if (lane == 0) vsump[row] = s + bvsum[0];
}

// K3: wvo[m,j] = sum_k vsump[m,k] * Wo[j,k]   (16x2048 @ 2048x2048^T, fp32 WMMA)
// One wave per 16-column output tile; B tile (16 Wo rows x KT) staged in LDS
// via async-to-LDS b128, double buffered on ASYNCcnt.
__global__ void wvo_gemm_kernel(const float* __restrict__ vsump, // [16, EMB], rows 4..15 = 0
                                const float* __restrict__ Wo,    // [EMB, EMB]
                                float* __restrict__ wvo) {       // [NBATCH, EMB]
  __shared__ float bt[WPB][2][16][KTP];
  const int wid  = threadIdx.x >> 5;
  const int lane = threadIdx.x & 31;
  const int wave = blockIdx.x * WPB + wid;
  const int ncol = lane & 15;   // output column within tile / B row / A row
  const int hi   = lane >> 4;   // lane half selects K pair {2hi, 2hi+1} per step
  const int j0   = wave * 16;

#if DEVICE_FAST
  // Issue one 16xKT B tile as 8 async b128 wave-instructions (512B each).
  auto issue = [&](int t, int buf) {
    const int kt0 = t * KT;
    #pragma unroll
    for (int it = 0; it < 8; ++it) {
      const int chunk = it * 32 + lane;        // 0..255 b128 chunks
      const int r = chunk >> 4;                // Wo row within tile
      const int c = (chunk & 15) << 2;         // float offset within row
      const float* g = Wo + (size_t)(j0 + r) * EMB + kt0 + c;
      const unsigned la = (unsigned)(uintptr_t)&bt[wid][buf][r][c]; // LDS byte offset
      asm volatile("global_load_async_to_lds_b128 %0, %1, off"
                   :: "v"(la), "v"(g) : "memory");
    }
  };

  v8f acc = {};
  const float* __restrict__ arow = vsump + (size_t)ncol * EMB;  // zero-padded rows
  issue(0, 0);
  #pragma unroll 1
  for (int t = 0; t < EMB / KT; ++t) {
    const int buf = t & 1;
    if (t + 1 < EMB / KT) {
      issue(t + 1, buf ^ 1);
      asm volatile("s_wait_asynccnt 0x8" ::: "memory"); // tile t landed, t+1 in flight
    } else {
      asm volatile("s_wait_asynccnt 0x0" ::: "memory");
    }
    const float* __restrict__ bl = &bt[wid][buf][ncol][0];
    const v2f* __restrict__ ap = (const v2f*)(arow + t * KT);
    #pragma unroll
    for (int kk = 0; kk < KT; kk += 4) {
      v2f a = ap[(kk >> 1) + hi];                    // A[ncol][K=base+2hi..+1]
      v2f b = *(const v2f*)(bl + kk + hi * 2);       // B[K][ncol] from LDS
      acc = __builtin_amdgcn_wmma_f32_16x16x4_f32(false, a, false, b,
                                                  (short)0, acc, false, false);
    }
  }
  // C/D layout: VGPR m -> row M=m on lanes 0-15; only M<4 carry real data.
  if (hi == 0) {
    #pragma unroll
    for (int m = 0; m < NBATCH; ++m)
      wvo[(size_t)m * EMB + j0 + ncol] = acc[m];
  }
#else
  // Host pass / non-gfx1250 fallback: 2 output rows per lane, plain FMA.
  float acc0 = 0.f, acc1 = 0.f;
  const int m0 = hi * 2;
  const float* __restrict__ worow = Wo + (size_t)(j0 + ncol) * EMB;
  const float* __restrict__ v0 = vsump + (size_t)m0 * EMB;
  const float* __restrict__ v1 = vsump + (size_t)(m0 + 1) * EMB;
  for (int k = 0; k < EMB; ++k) {
    const float w = worow[k];
    acc0 += v0[k] * w;
    acc1 += v1[k] * w;
  }
  wvo[(size_t)m0 * EMB + j0 + ncol] = acc0;
  wvo[(size_t)(m0 + 1) * EMB + j0 + ncol] = acc1;
#endif
}

// K4: out[n,q,j] = wvo[n,j] + bo[j], identical for every q.
// Streaming b128 NT stores (write-once 67 MB; keep it out of L2's working set).
__global__ void broadcast_out_kernel(const float* __restrict__ wvo,
                                     const float* __restrict__ bo,
                                     float* __restrict__ out) {
  const int n  = blockIdx.z;
  const int j4 = blockIdx.x * blockDim.x + threadIdx.x;  // float4 index, 0..EMB/4-1
  const int q0 = blockIdx.y * 32;
  const v4f* __restrict__ wv4 = (const v4f*)(wvo + (size_t)n * EMB);
  const v4f* __restrict__ bo4 = (const v4f*)bo;
  const v4f val = wv4[j4] + bo4[j4];
  v4f* outp = (v4f*)(out + ((size_t)n * LSEQ + q0) * EMB) + j4;
  #pragma unroll 4
  for (int q = 0; q < 32; ++q) {
    __builtin_nontemporal_store(val, outp);
    outp += EMB / 4;
  }
}

extern "C" void kernel_launch(void* const* d_in, const int* in_sizes, int n_in,
                              void* d_out, int out_size, void* d_ws, size_t ws_size,
                              hipStream_t stream) {
  (void)in_sizes; (void)n_in; (void)out_size; (void)ws_size;
  // setup_inputs order: values, keys, queries, Wv, bv, Wk, bk, Wq, bq, Wo, bo
  const float* values = (const float*)d_in[0];
  const float* Wv     = (const float*)d_in[3];
  const float* bv     = (const float*)d_in[4];
  const float* Wo     = (const float*)d_in[9];
  const float* bo     = (const float*)d_in[10];
  float* out = (float*)d_out;

  // Workspace layout (256B aligned):
  //   wcol  : EMB floats
  //   bvsum : 1 float (256B slot)
  //   vsump : 16*EMB floats (zero-padded A matrix; rows 0..3 = per-batch vsum)
  //   wvo   : NBATCH*EMB floats
  char* ws = (char*)d_ws;
  float* wcol  = (float*)(ws);
  float* bvsum = (float*)(ws + 8192);
  float* vsump = (float*)(ws + 8448);
  float* wvo   = (float*)(ws + 8448 + 16 * EMB * 4);

  pad_zero_kernel<<<dim3((16 - NBATCH) * EMB / 256), dim3(256), 0, stream>>>(vsump);
  colsum_bv_kernel<<<dim3(EMB / 256), dim3(256), 0, stream>>>(Wv, bv, wcol, bvsum);
  vsum_kernel<<<dim3((NBATCH * LSEQ) / 8), dim3(256), 0, stream>>>(values, wcol, bvsum, vsump);
  wvo_gemm_kernel<<<dim3(EMB / 16 / WPB), dim3(WPB * 32), 0, stream>>>(vsump, Wo, wvo);
  broadcast_out_kernel<<<dim3(EMB / (256 * 4), LSEQ / 32, NBATCH), dim3(256), 0, stream>>>(
      wvo, bo, out);
}